// DistanceBasedLogitLoss_36112085025079
// MI455X (gfx1250) — compile-verified
//
#include <hip/hip_runtime.h>
#include <math.h>

// ---------------------------------------------------------------------------
// MI455X / gfx1250, wave32. All matrix math goes through v_wmma_f32_16x16x4_f32
// (full-f32 WMMA: pairwise distances cancel catastrophically in low precision).
// FFT is done as DFT-by-GEMM using the symmetry of F:  ff = F * (r * F).
// Pass 1: B1 = r*F  (stored transposed so pass 2's B-fragments are contiguous)
// Pass 2: ff = F*B1, psd += |ff|^2
// Every inner-loop fragment load is a contiguous 64-bit global load.
// ---------------------------------------------------------------------------

typedef __attribute__((ext_vector_type(2))) float v2f;
typedef __attribute__((ext_vector_type(4))) float v4f;
typedef __attribute__((ext_vector_type(8))) float v8f;

#define N_SAMP 256
#define HWDIM  320
#define DDIM   102400      // 320*320
#define KSPLIT 25
#define KCHUNK 4096        // DDIM / KSPLIT

__device__ __forceinline__ v8f wmma_f32(v2f a, v2f b, v8f c) {
  // (neg_a, A, neg_b, B, c_mod, C, reuse_a, reuse_b)
  return __builtin_amdgcn_wmma_f32_16x16x4_f32(false, a, false, b, (short)0, c,
                                               false, false);
}

// ------------------------------- utilities ---------------------------------

__global__ __launch_bounds__(256) void zero_f32(float* __restrict__ p, int n) {
  int i = blockIdx.x * 256 + threadIdx.x;
  if (i < n) p[i] = 0.0f;
}

// DFT twiddle matrices Fre/Fim, 320x320. (k*j) mod 320 keeps the argument small.
__global__ __launch_bounds__(256) void init_dft(float* __restrict__ Fre,
                                                float* __restrict__ Fim) {
  int j = blockIdx.x * 16 + threadIdx.x;
  int k = blockIdx.y * 16 + threadIdx.y;
  if (k < HWDIM && j < HWDIM) {
    int t = (k * j) % HWDIM;
    float ang = -2.0f * 3.14159265358979323846f * (float)t / (float)HWDIM;
    Fre[k * HWDIM + j] = cosf(ang);
    Fim[k * HWDIM + j] = sinf(ang);
  }
}

// Per-row sum and sum-of-squares of X [256 x 102400].
__global__ __launch_bounds__(256) void rowsum_kernel(const float* __restrict__ X,
                                                     float* __restrict__ s,
                                                     float* __restrict__ sq) {
  __shared__ float sh1[256];
  __shared__ float sh2[256];
  int row = blockIdx.x;
  const float* x = X + (size_t)row * DDIM;
  float a = 0.f, b = 0.f;
  for (int i = threadIdx.x; i < DDIM; i += 256) {
    float v = x[i];
    a += v;
    b += v * v;
  }
  sh1[threadIdx.x] = a;
  sh2[threadIdx.x] = b;
  __syncthreads();
  for (int off = 128; off > 0; off >>= 1) {
    if ((int)threadIdx.x < off) {
      sh1[threadIdx.x] += sh1[threadIdx.x + off];
      sh2[threadIdx.x] += sh2[threadIdx.x + off];
    }
    __syncthreads();
  }
  if (threadIdx.x == 0) {
    s[row] = sh1[0];
    sq[row] = sh2[0];
  }
}

// ------------------------------ gram = X X^T -------------------------------
// One wave per 16x16 output tile, split-K over gridDim.z, f32 atomics into gram.
// A-frag: lane l -> A[l&15][2*(l>>4)+{0,1}]; B-frag: lane l -> B[2*(l>>4)+{0,1}][l&15].
__global__ __launch_bounds__(32) void gram_kernel(const float* __restrict__ X,
                                                  float* __restrict__ gram) {
  const int tm = blockIdx.x * 16;
  const int tn = blockIdx.y * 16;
  const int lane = threadIdx.x;
  const int mn = lane & 15;          // A-row / B-col owned by this lane
  const int kb = (lane >> 4) * 2;    // K sub-offset within the 4-wide step
  const int K0 = blockIdx.z * KCHUNK;
  const int K1 = K0 + KCHUNK;

  const float* Arow = X + (size_t)(tm + mn) * DDIM;
  const float* Brow = X + (size_t)(tn + mn) * DDIM;  // B = X^T: B[k][n]=X[tn+n][k]

  v8f acc = {};
  for (int k = K0; k < K1; k += 64) {
    __builtin_prefetch(Arow + k + 512, 0, 1);
    __builtin_prefetch(Brow + k + 512, 0, 1);
#pragma unroll
    for (int kk = 0; kk < 64; kk += 4) {
      v2f a = *(const v2f*)(Arow + k + kk + kb);
      v2f b = *(const v2f*)(Brow + k + kk + kb);
      acc = wmma_f32(a, b, acc);
    }
  }
#pragma unroll
  for (int v = 0; v < 8; ++v) {
    int row = tm + v + 8 * (lane >> 4);
    int col = tn + mn;
    atomicAdd(&gram[row * N_SAMP + col], acc[v]);
  }
}

// ------------------------ FFT pass 1: B1 = r * F ---------------------------
// A = r (rows contiguous). B-frag uses F's symmetry: F[k][cn] = F[cn][k], so
// both fragments are contiguous v2f loads. Output stored TRANSPOSED
// (B1t[k2][j1]) so each lane's 8 accumulators are contiguous -> b128 stores,
// and pass 2's B-fragments become contiguous too.
__global__ __launch_bounds__(32) void fft_pass1(const float* __restrict__ R,
                                                const float* __restrict__ Fre,
                                                const float* __restrict__ Fim,
                                                float* __restrict__ B1re_t,
                                                float* __restrict__ B1im_t,
                                                int n0) {
  const int tm = blockIdx.x * 16;   // j1 block (rows of r)
  const int tn = blockIdx.y * 16;   // k2 block (output frequency columns)
  const int sl = blockIdx.z;
  const int lane = threadIdx.x;
  const int mn = lane & 15;
  const int kb = (lane >> 4) * 2;
  const int cn = tn + mn;

  const float* r = R + (size_t)(n0 + sl) * DDIM + (size_t)(tm + mn) * HWDIM;
  const float* FreCol = Fre + (size_t)cn * HWDIM;  // symmetric access
  const float* FimCol = Fim + (size_t)cn * HWDIM;

  v8f accRe = {}, accIm = {};
#pragma unroll 4
  for (int k = 0; k < HWDIM; k += 4) {
    v2f a   = *(const v2f*)(r + k + kb);
    v2f bre = *(const v2f*)(FreCol + k + kb);
    v2f bim = *(const v2f*)(FimCol + k + kb);
    accRe = wmma_f32(a, bre, accRe);
    accIm = wmma_f32(a, bim, accIm);
  }

  // Transposed store: address = cn*320 + (tm + 8*hi) + v  -> 8 contiguous floats
  const int hi = lane >> 4;
  float* ore = B1re_t + (size_t)sl * DDIM + (size_t)cn * HWDIM + tm + 8 * hi;
  float* oim = B1im_t + (size_t)sl * DDIM + (size_t)cn * HWDIM + tm + 8 * hi;
  v4f r0 = {accRe[0], accRe[1], accRe[2], accRe[3]};
  v4f r1 = {accRe[4], accRe[5], accRe[6], accRe[7]};
  v4f i0 = {accIm[0], accIm[1], accIm[2], accIm[3]};
  v4f i1 = {accIm[4], accIm[5], accIm[6], accIm[7]};
  *(v4f*)(ore)     = r0;
  *(v4f*)(ore + 4) = r1;
  *(v4f*)(oim)     = i0;
  *(v4f*)(oim + 4) = i1;
}

// ------------------- FFT pass 2: ff = F * B1, psd += |ff|^2 ----------------
// A-frags: F rows (contiguous). B-frags: B1[k][cn] = B1t[cn*320+k] (contiguous).
// ffre = Fre*B1re - Fim*B1im ; ffim = Fre*B1im + Fim*B1re.
__global__ __launch_bounds__(32) void fft_pass2(const float* __restrict__ B1re_t,
                                                const float* __restrict__ B1im_t,
                                                const float* __restrict__ Fre,
                                                const float* __restrict__ Fim,
                                                float* __restrict__ psd,
                                                int nslices) {
  const int tm = blockIdx.x * 16;   // k1 block
  const int tn = blockIdx.y * 16;   // k2 block
  const int z0 = blockIdx.z, nz = gridDim.z;
  const int lane = threadIdx.x;
  const int mn = lane & 15;
  const int kb = (lane >> 4) * 2;
  const int cn = tn + mn;

  const float* FreRow = Fre + (size_t)(tm + mn) * HWDIM;
  const float* FimRow = Fim + (size_t)(tm + mn) * HWDIM;

  float pacc[8] = {0.f, 0.f, 0.f, 0.f, 0.f, 0.f, 0.f, 0.f};

  for (int sl = z0; sl < nslices; sl += nz) {
    const float* bre = B1re_t + (size_t)sl * DDIM + (size_t)cn * HWDIM;
    const float* bim = B1im_t + (size_t)sl * DDIM + (size_t)cn * HWDIM;
    v8f cre = {}, cim = {};
#pragma unroll 4
    for (int k = 0; k < HWDIM; k += 4) {
      v2f afr = *(const v2f*)(FreRow + k + kb);
      v2f afi = *(const v2f*)(FimRow + k + kb);
      v2f br  = *(const v2f*)(bre + k + kb);
      v2f bi  = *(const v2f*)(bim + k + kb);
      v2f afin = -afi;  // F32 WMMA has no A-negate modifier (NEG is C-only)
      cre = wmma_f32(afr,  br, cre);   //  Fre*B1re
      cre = wmma_f32(afin, bi, cre);   // -Fim*B1im
      cim = wmma_f32(afr,  bi, cim);   //  Fre*B1im
      cim = wmma_f32(afi,  br, cim);   //  Fim*B1re
    }
#pragma unroll
    for (int v = 0; v < 8; ++v) pacc[v] += cre[v] * cre[v] + cim[v] * cim[v];
  }
#pragma unroll
  for (int v = 0; v < 8; ++v) {
    int row = tm + v + 8 * (lane >> 4);
    atomicAdd(&psd[row * HWDIM + cn], pacc[v]);
  }
}

// ------------------------------ distance loss ------------------------------
// One workgroup, thread j owns column j of the 256x256 distance matrix.
__global__ __launch_bounds__(256) void loss_kernel(const float* __restrict__ gram,
                                                   const float* __restrict__ s,
                                                   const float* __restrict__ sq,
                                                   float* __restrict__ out_scalar) {
  const float eps = 1e-6f;
  const float Df = (float)DDIM;
  const int j = threadIdx.x;
  const float sqj = sq[j], sj = s[j];
  const int g = j >> 2;

  float posacc = 0.f;  // colsum+rowsum of positive matrix at j = sum over
                       // same-group partners of -dist
  float Sj = 0.f;      // this thread's share of the upper-triangle total
  for (int i = 0; i < N_SAMP; ++i) {
    if (i == j) continue;
    float d2 = sq[i] + sqj - 2.f * gram[i * N_SAMP + j] +
               2.f * eps * (s[i] - sj) + Df * eps * eps;
    float dist = sqrtf(fmaxf(d2, 0.f));
    if ((i >> 2) == g) posacc -= dist;
    if (i < j) Sj -= dist;
  }

  __shared__ float sh[256];
  sh[j] = Sj;
  __syncthreads();
  for (int off = 128; off > 0; off >>= 1) {
    if (j < off) sh[j] += sh[j + off];
    __syncthreads();
  }
  float S = sh[0];
  __syncthreads();
  float lossj = -logf(posacc / S);
  sh[j] = lossj;
  __syncthreads();
  for (int off = 128; off > 0; off >>= 1) {
    if (j < off) sh[j] += sh[j + off];
    __syncthreads();
  }
  if (j == 0) out_scalar[0] = sh[0];
}

// ------------------------- PSD regularizer + combine -----------------------
__global__ __launch_bounds__(1024) void final_kernel(const float* __restrict__ psd_sum,
                                                     const float* __restrict__ loss_all,
                                                     float* __restrict__ out) {
  const float invN = 1.0f / (float)N_SAMP;
  float slog = 0.f, ssum = 0.f;
  for (int i = threadIdx.x; i < DDIM; i += 1024) {
    float p = psd_sum[i] * invN;
    slog += logf(p);
    ssum += p;
  }
  __shared__ float sa[1024];
  __shared__ float sb[1024];
  sa[threadIdx.x] = slog;
  sb[threadIdx.x] = ssum;
  __syncthreads();
  for (int off = 512; off > 0; off >>= 1) {
    if ((int)threadIdx.x < off) {
      sa[threadIdx.x] += sa[threadIdx.x + off];
      sb[threadIdx.x] += sb[threadIdx.x + off];
    }
    __syncthreads();
  }
  if (threadIdx.x == 0) {
    float reg = sa[0] / (float)DDIM - logf(sb[0] / (float)DDIM);
    out[0] = loss_all[0] - 0.1f * reg;
  }
}

// ------------------------------- orchestration -----------------------------
// Workspace layout (float offsets):
//   0       Fre  (102400)
//   102400  Fim  (102400)
//   204800  gram (65536)
//   270336  s    (256)
//   270592  sq   (256)
//   270848  psd  (102400)
//   373248  scalars (16)
//   373264  batch area: B1re_t[nb*102400], B1im_t[nb*102400]
extern "C" void kernel_launch(void* const* d_in, const int* in_sizes, int n_in,
                              void* d_out, int out_size, void* d_ws, size_t ws_size,
                              hipStream_t stream) {
  (void)in_sizes; (void)n_in; (void)out_size;
  const float* X = (const float*)d_in[0];
  float* w = (float*)d_ws;

  float* Fre  = w;
  float* Fim  = w + 102400;
  float* gram = w + 204800;
  float* svec = w + 270336;
  float* sqv  = w + 270592;
  float* psd  = w + 270848;
  float* scal = w + 373248;
  float* batch = w + 373264;

  init_dft<<<dim3(20, 20), dim3(16, 16), 0, stream>>>(Fre, Fim);

  {
    int nz = 373264 - 204800;  // gram + s + sq + psd + scalars
    zero_f32<<<(nz + 255) / 256, 256, 0, stream>>>(gram, nz);
  }

  rowsum_kernel<<<N_SAMP, 256, 0, stream>>>(X, svec, sqv);
  gram_kernel<<<dim3(16, 16, KSPLIT), 32, 0, stream>>>(X, gram);

  // Slice batching sized from ws_size (deterministic for fixed ws_size).
  size_t wsf = ws_size / sizeof(float);
  size_t avail = (wsf > 373264) ? (wsf - 373264) : 0;
  int nb = (int)(avail / (2u * (size_t)DDIM));
  if (nb < 1) nb = 1;
  if (nb > N_SAMP) nb = N_SAMP;
  float* B1re = batch;
  float* B1im = batch + (size_t)nb * DDIM;

  for (int n0 = 0; n0 < N_SAMP; n0 += nb) {
    int cnt = (N_SAMP - n0 < nb) ? (N_SAMP - n0) : nb;
    fft_pass1<<<dim3(20, 20, cnt), 32, 0, stream>>>(X, Fre, Fim, B1re, B1im, n0);
    fft_pass2<<<dim3(20, 20, 16), 32, 0, stream>>>(B1re, B1im, Fre, Fim, psd, cnt);
  }

  loss_kernel<<<1, 256, 0, stream>>>(gram, svec, sqv, scal);
  final_kernel<<<1, 1024, 0, stream>>>(psd, scal, (float*)d_out);
}